// PoolClusters_14139032339209
// MI455X (gfx1250) — compile-verified
//
#include <hip/hip_runtime.h>
#include <hip/hip_bf16.h>

// Segment-mean: out[c,:] = mean of rows of x whose sl[i]==c.
// Bandwidth-bound (~1GB read => ~45us floor at 23.3 TB/s).
// Strategy: counting-sort rows by cluster id, then per-cluster gather-sum
// with fully coalesced 128-bit read-once (non-temporal) loads, no float atomics.

typedef float v4f __attribute__((ext_vector_type(4)));  // native vec for NT/b128 ops

// ---------------------------------------------------------------------------
// Path A: counting sort + gather  (needs ws >= (3*C + N) ints)
// ---------------------------------------------------------------------------

__global__ void zero_i32_kernel(int* __restrict__ p, int n) {
    int i = blockIdx.x * blockDim.x + threadIdx.x;
    if (i < n) p[i] = 0;
}

// LDS-privatized histogram of cluster ids.
__global__ __launch_bounds__(256) void hist_kernel(const int* __restrict__ sl, int n,
                                                   int* __restrict__ counts, int C) {
    __shared__ int h[4096];
    for (int i = threadIdx.x; i < C; i += blockDim.x) h[i] = 0;
    __syncthreads();
    long long stride = (long long)gridDim.x * blockDim.x;
    for (long long i = (long long)blockIdx.x * blockDim.x + threadIdx.x; i < n; i += stride) {
        atomicAdd(&h[sl[i]], 1);            // ds_add_u32 (LDS atomic)
    }
    __syncthreads();
    for (int i = threadIdx.x; i < C; i += blockDim.x) {
        int v = h[i];
        if (v) atomicAdd(&counts[i], v);    // few global atomics per block
    }
}

// Exclusive scan of counts[0..C) -> offsets, single block of 1024 (C <= 4096).
// Hierarchical wave32 scan (no long serial dependency chain).
__global__ __launch_bounds__(1024) void scan_kernel(const int* __restrict__ counts,
                                                    int* __restrict__ offsets, int C) {
    __shared__ int wsum[32];
    int t = threadIdx.x;
    int base = t * 4;
    int c0 = (base + 0 < C) ? counts[base + 0] : 0;
    int c1 = (base + 1 < C) ? counts[base + 1] : 0;
    int c2 = (base + 2 < C) ? counts[base + 2] : 0;
    int c3 = (base + 3 < C) ? counts[base + 3] : 0;
    int l0 = 0, l1 = c0, l2 = c0 + c1, l3 = c0 + c1 + c2;
    int s  = c0 + c1 + c2 + c3;            // group total (4 bins per thread)

    // inclusive scan of group totals within each wave32
    int v = s;
    #pragma unroll
    for (int d = 1; d < 32; d <<= 1) {
        int u = __shfl_up(v, d, 32);
        if ((t & 31) >= d) v += u;
    }
    if ((t & 31) == 31) wsum[t >> 5] = v;   // wave totals
    __syncthreads();
    if (t < 32) {                            // scan the 32 wave totals (wave 0)
        int w = wsum[t];
        #pragma unroll
        for (int d = 1; d < 32; d <<= 1) {
            int u = __shfl_up(w, d, 32);
            if (t >= d) w += u;
        }
        wsum[t] = w;                         // inclusive
    }
    __syncthreads();
    int waveExcl = (t >> 5) ? wsum[(t >> 5) - 1] : 0;
    int excl = waveExcl + (v - s);           // exclusive scan of group totals
    if (base + 0 < C) offsets[base + 0] = excl + l0;
    if (base + 1 < C) offsets[base + 1] = excl + l1;
    if (base + 2 < C) offsets[base + 2] = excl + l2;
    if (base + 3 < C) offsets[base + 3] = excl + l3;
}

// Scatter row indices into cluster-sorted order.
__global__ __launch_bounds__(256) void scatter_kernel(const int* __restrict__ sl, int n,
                                                      const int* __restrict__ offsets,
                                                      int* __restrict__ cursors,
                                                      int* __restrict__ perm) {
    long long stride = (long long)gridDim.x * blockDim.x;
    for (long long i = (long long)blockIdx.x * blockDim.x + threadIdx.x; i < n; i += stride) {
        int c = sl[i];
        int p = atomicAdd(&cursors[c], 1);
        perm[offsets[c] + p] = (int)i;
    }
}

// One block per cluster. 256 threads = 64 float4-columns x 4 row-slots:
// each inner step covers 4 rows with global_load_b128 (non-temporal, read-once),
// unrolled x2 so each thread keeps 2 b128 loads in flight; each wave's request
// is 512B contiguous. Final 4-way cross-slot reduction via 4KB of LDS.
__global__ __launch_bounds__(256) void gather_kernel(const float* __restrict__ x,
                                                     const int* __restrict__ perm,
                                                     const int* __restrict__ offsets,
                                                     const int* __restrict__ counts,
                                                     float* __restrict__ out) {
    const int c    = blockIdx.x;
    const int t    = threadIdx.x;
    const int col4 = t & 63;    // which float4 within the 256-float row
    const int rgrp = t >> 6;    // row slot 0..3
    const int beg  = offsets[c];
    const int cnt  = counts[c];
    const v4f* __restrict__ x4 = (const v4f*)x;   // row r = x4[r*64 .. r*64+63]

    __shared__ int rows[256];
    __shared__ v4f partial[4][64];
    v4f acc = {0.0f, 0.0f, 0.0f, 0.0f};

    for (int baseI = 0; baseI < cnt; baseI += 256) {
        int m = min(256, cnt - baseI);
        __syncthreads();
        if (t < m) rows[t] = perm[beg + baseI + t];
        __syncthreads();
        int j = 0;
        for (; j + 8 <= m; j += 8) {
            int row0 = rows[j + rgrp];
            int row1 = rows[j + 4 + rgrp];
            if (j + 24 < m) {
                // global_prefetch_b8 look-ahead into L2
                __builtin_prefetch(&x4[(size_t)rows[j + 16 + rgrp] * 64 + col4], 0, 0);
            }
            v4f v0 = __builtin_nontemporal_load(&x4[(size_t)row0 * 64 + col4]);
            v4f v1 = __builtin_nontemporal_load(&x4[(size_t)row1 * 64 + col4]);
            acc += v0;
            acc += v1;
        }
        for (; j < m; j += 4) {
            int r = j + rgrp;
            if (r < m) {
                acc += __builtin_nontemporal_load(&x4[(size_t)rows[r] * 64 + col4]);
            }
        }
    }

    partial[rgrp][col4] = acc;
    __syncthreads();
    if (rgrp == 0) {
        v4f s = partial[0][col4] + partial[1][col4] + partial[2][col4] + partial[3][col4];
        float den = fmaxf((float)cnt, 1.0f);
        v4f res;
        res.x = s.x / den;
        res.y = s.y / den;
        res.z = s.z / den;
        res.w = s.w / den;
        ((v4f*)out)[(size_t)c * 64 + col4] = res;   // b128 coalesced store
    }
}

// ---------------------------------------------------------------------------
// Path B: vectorized atomic scatter-add (fallback if ws too small for perm)
// ---------------------------------------------------------------------------

__global__ void initB_kernel(float* __restrict__ out, float* __restrict__ counts,
                             long long total_out, int C) {
    long long i = (long long)blockIdx.x * blockDim.x + threadIdx.x;
    if (i < total_out) out[i] = 0.0f;
    if (i < C) counts[i] = 0.0f;
}

__global__ __launch_bounds__(256) void accB_kernel(const v4f* __restrict__ x4,
                                                   const int* __restrict__ sl,
                                                   float* __restrict__ out,
                                                   float* __restrict__ counts, int n) {
    long long total  = (long long)n * 64;       // D==256 -> 64 float4 per row
    long long stride = (long long)gridDim.x * blockDim.x;
    for (long long i = (long long)blockIdx.x * blockDim.x + threadIdx.x; i < total; i += stride) {
        int row = (int)(i >> 6);
        int col = (int)(i & 63);
        int c = sl[row];
        v4f v = __builtin_nontemporal_load(&x4[i]);
        float* dst = out + (size_t)c * 256 + col * 4;
        atomicAdd(dst + 0, v.x);
        atomicAdd(dst + 1, v.y);
        atomicAdd(dst + 2, v.z);
        atomicAdd(dst + 3, v.w);
        if (col == 0) atomicAdd(counts + c, 1.0f);
    }
}

__global__ void accG_kernel(const float* __restrict__ x, const int* __restrict__ sl,
                            float* __restrict__ out, float* __restrict__ counts,
                            int n, int D) {
    long long total  = (long long)n * D;
    long long stride = (long long)gridDim.x * blockDim.x;
    for (long long i = (long long)blockIdx.x * blockDim.x + threadIdx.x; i < total; i += stride) {
        int row = (int)(i / D);
        int d   = (int)(i % D);
        int c = sl[row];
        atomicAdd(out + (size_t)c * D + d, x[i]);
        if (d == 0) atomicAdd(counts + c, 1.0f);
    }
}

__global__ void finalB_kernel(float* __restrict__ out, const float* __restrict__ counts,
                              long long total, int D) {
    long long i = (long long)blockIdx.x * blockDim.x + threadIdx.x;
    if (i < total) out[i] /= fmaxf(counts[i / D], 1.0f);
}

// ---------------------------------------------------------------------------

extern "C" void kernel_launch(void* const* d_in, const int* in_sizes, int n_in,
                              void* d_out, int out_size, void* d_ws, size_t ws_size,
                              hipStream_t stream) {
    const float* x  = (const float*)d_in[0];
    const int*   sl = (const int*)d_in[1];
    const int n = in_sizes[1];
    const int D = in_sizes[0] / n;
    const int C = out_size / D;
    float* out = (float*)d_out;

    const size_t needA = ((size_t)3 * C + (size_t)n) * sizeof(int);
    const bool canA = (D == 256) && (C <= 4096) && (ws_size >= needA);

    if (canA) {
        int* counts  = (int*)d_ws;
        int* cursors = counts + C;
        int* offsets = counts + 2 * C;
        int* perm    = counts + 3 * C;
        zero_i32_kernel<<<(2 * C + 255) / 256, 256, 0, stream>>>(counts, 2 * C);
        hist_kernel<<<1024, 256, 0, stream>>>(sl, n, counts, C);
        scan_kernel<<<1, 1024, 0, stream>>>(counts, offsets, C);
        scatter_kernel<<<2048, 256, 0, stream>>>(sl, n, offsets, cursors, perm);
        gather_kernel<<<C, 256, 0, stream>>>(x, perm, offsets, counts, out);
    } else if (ws_size >= (size_t)C * sizeof(float)) {
        float* counts = (float*)d_ws;
        long long total = (long long)C * D;
        long long initN = total > C ? total : C;
        initB_kernel<<<(unsigned)((initN + 255) / 256), 256, 0, stream>>>(out, counts, total, C);
        if (D == 256) {
            accB_kernel<<<2048, 256, 0, stream>>>((const v4f*)x, sl, out, counts, n);
        } else {
            accG_kernel<<<2048, 256, 0, stream>>>(x, sl, out, counts, n, D);
        }
        finalB_kernel<<<(unsigned)((total + 255) / 256), 256, 0, stream>>>(out, counts, total, D);
    }
}